// SVPFModel_9148280340772
// MI455X (gfx1250) — compile-verified
//
#include <hip/hip_runtime.h>
#include <hip/hip_bf16.h>
#include <math.h>

// ---------------- problem constants ----------------
#define Bn 8
#define Nn 2048
#define NTILE (Nn / 16)      // 128 j/i tiles
#define NB 1024              // histogram bins per level
#define KRANK 2097151LL      // (N*N - 1) / 2, 0-based lower median rank
#define A_C 0.9f
#define HALF_A2 0.405f       // 0.5 * A^2 / sigma^2
#define LOGN 7.6246190f      // log(2048)
#define ALPHA_C 0.9f
#define LR_C 0.1f
#define EPS_C 1e-8f

typedef __attribute__((ext_vector_type(2))) float v2f;
typedef __attribute__((ext_vector_type(8))) float v8f;

// V_WMMA_F32_16X16X4_F32 : D(16x16,f32) = A(16x4,f32) x B(4x16,f32) + C
// A: lane L holds row M=L&15 ; v0,v1 = K (L<16 ? {0,1} : {2,3})
// D: lane L holds N=L&15 ; vgpr v holds M = v + 8*(L>>4)
__device__ __forceinline__ v8f wmma4(v2f a, v2f b) {
    v8f c = {0.f, 0.f, 0.f, 0.f, 0.f, 0.f, 0.f, 0.f};
    return __builtin_amdgcn_wmma_f32_16x16x4_f32(
        false, a, false, b, (short)0, c, false, false);
}

__device__ __forceinline__ v2f ld2(const float* p) { return *(const v2f*)p; }

// ---------------- K0: init x = A*prev + noise, x2, p2, step=0 --------------
__global__ void k_init(const float* __restrict__ prev, const float* __restrict__ noise,
                       float* __restrict__ x0, float* __restrict__ x2_0,
                       float* __restrict__ p2, float* __restrict__ step) {
    int t = blockIdx.x * blockDim.x + threadIdx.x;
    if (t >= Bn * Nn) return;
    float4 p = ((const float4*)prev)[t];
    float4 nz = ((const float4*)noise)[t];
    float4 xv;
    xv.x = A_C * p.x + nz.x; xv.y = A_C * p.y + nz.y;
    xv.z = A_C * p.z + nz.z; xv.w = A_C * p.w + nz.w;
    ((float4*)x0)[t] = xv;
    x2_0[t] = xv.x * xv.x + xv.y * xv.y + xv.z * xv.z + xv.w * xv.w;
    p2[t]   = p.x * p.x + p.y * p.y + p.z * p.z + p.w * p.w;
    float4 z = {0.f, 0.f, 0.f, 0.f};
    ((float4*)step)[t] = z;
}

// ---------------- Kprep: per-batch sq-distance bound, zero hist ------------
// Median selection runs in SQUARED-distance space (sqrt is monotonic, and the
// reference only needs med^2 for the bandwidth) -> no sqrt in hot loops.
__global__ void k_prep(const float* __restrict__ x2c, float* __restrict__ medinfo,
                       int* __restrict__ hist) {
    __shared__ float red[32];
    int b = blockIdx.x, t = threadIdx.x;
    hist[b * NB + t] = 0;
    float m = fmaxf(x2c[b * Nn + t], x2c[b * Nn + t + 1024]);
    for (int o = 16; o > 0; o >>= 1) m = fmaxf(m, __shfl_xor(m, o, 32));
    if ((t & 31) == 0) red[t >> 5] = m;
    __syncthreads();
    if (t < 32) {
        float mm = red[t];
        for (int o = 16; o > 0; o >>= 1) mm = fmaxf(mm, __shfl_xor(mm, o, 32));
        if (t == 0) {
            medinfo[b * 4 + 0] = 0.f;
            medinfo[b * 4 + 1] = 4.f * mm + 1e-6f;  // ||xi-xj||^2 <= 4*max||x||^2
        }
    }
}

// ---------------- Khist: WMMA Gram tiles -> sq-distance histogram ----------
// fine==0: coarse pass, every sq is in [lo,hi) by construction -> branchless.
// fine==1: refinement pass, predicate pays (most elements out of range).
__global__ void k_hist(const float* __restrict__ x, const float* __restrict__ x2,
                       const float* __restrict__ medinfo, int* __restrict__ hist,
                       int fine) {
    __shared__ int lh[NB];
    int b = blockIdx.x >> 5, grp = blockIdx.x & 31;
    int lane = threadIdx.x & 31, wv = threadIdx.x >> 5;
    for (int v = threadIdx.x; v < NB; v += blockDim.x) lh[v] = 0;
    __syncthreads();
    const float* xb = x + (size_t)b * Nn * 4;
    const float* x2b = x2 + (size_t)b * Nn;
    float lo = medinfo[b * 4 + 0], hi = medinfo[b * 4 + 1];
    float invw = (float)NB / (hi - lo);
    int itile = grp * 4 + wv;
    int ir = itile * 16 + (lane & 15);
    int ko = (lane >> 4) * 2;
    v2f bm = ld2(xb + ir * 4 + ko);
    float x2i = x2b[ir];
    if (!fine) {
        for (int J = 0; J < NTILE; ++J) {
            int jr = J * 16 + (lane & 15);
            __builtin_prefetch(xb + (J + 1) * 64, 0, 3);
            v2f am = ld2(xb + jr * 4 + ko);
            v8f d = wmma4(am, bm);
            int jb = J * 16 + ((lane >> 4) << 3);
            float4 q0 = *(const float4*)(x2b + jb);
            float4 q1 = *(const float4*)(x2b + jb + 4);
            float x2j[8] = {q0.x, q0.y, q0.z, q0.w, q1.x, q1.y, q1.z, q1.w};
#pragma unroll
            for (int u = 0; u < 8; ++u) {
                float sq = fmaxf(x2j[u] + x2i - 2.f * d[u], 0.f);
                int bin = (int)(sq * invw);
                bin = bin < NB - 1 ? bin : NB - 1;
                atomicAdd(&lh[bin], 1);   // always in range: no predicate
            }
        }
    } else {
        for (int J = 0; J < NTILE; ++J) {
            int jr = J * 16 + (lane & 15);
            __builtin_prefetch(xb + (J + 1) * 64, 0, 3);
            v2f am = ld2(xb + jr * 4 + ko);
            v8f d = wmma4(am, bm);
            int jb = J * 16 + ((lane >> 4) << 3);
            float4 q0 = *(const float4*)(x2b + jb);
            float4 q1 = *(const float4*)(x2b + jb + 4);
            float x2j[8] = {q0.x, q0.y, q0.z, q0.w, q1.x, q1.y, q1.z, q1.w};
#pragma unroll
            for (int u = 0; u < 8; ++u) {
                float sq = fmaxf(x2j[u] + x2i - 2.f * d[u], 0.f);
                if (sq >= lo && sq < hi) {
                    int bin = (int)((sq - lo) * invw);
                    bin = bin < NB - 1 ? bin : NB - 1;
                    atomicAdd(&lh[bin], 1);
                }
            }
        }
    }
    __syncthreads();
    for (int v = threadIdx.x; v < NB; v += blockDim.x) {
        int c = lh[v];
        if (c) atomicAdd(&hist[b * NB + v], c);
    }
}

// ---------------- Kscan: select rank bin, refine or finalize h -------------
__global__ void k_scan(int* __restrict__ hist, float* __restrict__ medinfo,
                       float* __restrict__ hval, int coarse) {
    __shared__ int sh[NB];
    int b = blockIdx.x, t = threadIdx.x;
    sh[t] = hist[b * NB + t];
    hist[b * NB + t] = 0;
    __syncthreads();
    if (t == 0) {
        float lo = medinfo[b * 4 + 0], hi = medinfo[b * 4 + 1];
        float wbin = (hi - lo) / (float)NB;
        long long k = coarse ? KRANK : (long long)medinfo[b * 4 + 2];
        long long c = 0;
        int tb = NB - 1;
        for (int i = 0; i < NB; ++i) {
            long long nc = c + (long long)sh[i];
            if (k < nc) { tb = i; break; }
            c = nc;
        }
        if (coarse) {
            medinfo[b * 4 + 0] = lo + (float)tb * wbin;
            medinfo[b * 4 + 1] = lo + (float)(tb + 1) * wbin;
            medinfo[b * 4 + 2] = (float)(k - c);
        } else {
            float med_sq = lo + ((float)tb + 0.5f) * wbin;  // == med^2
            float h = med_sq / LOGN;
            hval[b * 4 + 0] = h;
            hval[b * 4 + 1] = 1.f / h;
            hval[b * 4 + 2] = 2.f / h;
        }
    }
}

// ---------------- Kgrad: posterior score via WMMA + online softmax ---------
__global__ void k_grad(const float* __restrict__ x, const float* __restrict__ prev,
                       const float* __restrict__ p2, const float* __restrict__ obs,
                       float* __restrict__ g) {
    int b = blockIdx.x >> 5, grp = blockIdx.x & 31;
    int lane = threadIdx.x & 31, wv = threadIdx.x >> 5;
    int itile = grp * 4 + wv;
    const float* xb = x + (size_t)b * Nn * 4;
    const float* pb = prev + (size_t)b * Nn * 4;
    const float* p2b = p2 + (size_t)b * Nn;
    int il = itile * 16 + (lane & 15);
    int ko = (lane >> 4) * 2;
    v2f bm = ld2(xb + il * 4 + ko);
    float Mx = -3.0e38f, S = 0.f, V0 = 0.f, V1 = 0.f, V2 = 0.f, V3 = 0.f;
    for (int J = 0; J < NTILE; ++J) {
        int jr = J * 16 + (lane & 15);
        __builtin_prefetch(pb + (J + 1) * 64, 0, 3);
        v2f am = ld2(pb + jr * 4 + ko);          // A rows = prev[j-tile]
        v8f d = wmma4(am, bm);                   // d[u] = prev_j . x_i  (transposed tile)
        int jb = J * 16 + ((lane >> 4) << 3);
        float4 q0 = *(const float4*)(p2b + jb);
        float4 q1 = *(const float4*)(p2b + jb + 4);
        float p2j[8] = {q0.x, q0.y, q0.z, q0.w, q1.x, q1.y, q1.z, q1.w};
#pragma unroll
        for (int u = 0; u < 8; ++u) {
            int j = jb + u;
            float tl = A_C * d[u] - HALF_A2 * p2j[u];   // softmax logits (row consts cancel)
            // branchless online softmax: no exec-mask divergence
            float nM = fmaxf(Mx, tl);
            float sc = __expf(Mx - nM);   // 1 when max unchanged, 0 on first iter
            float w  = __expf(tl - nM);
            float4 pj = *(const float4*)(pb + j * 4);
            S  = S  * sc + w;
            V0 = V0 * sc + w * pj.x; V1 = V1 * sc + w * pj.y;
            V2 = V2 * sc + w * pj.z; V3 = V3 * sc + w * pj.w;
            Mx = nM;
        }
    }
    // merge lane-halves (same i, disjoint j sets)
    float oM = __shfl_xor(Mx, 16, 32), oS = __shfl_xor(S, 16, 32);
    float o0 = __shfl_xor(V0, 16, 32), o1 = __shfl_xor(V1, 16, 32);
    float o2 = __shfl_xor(V2, 16, 32), o3 = __shfl_xor(V3, 16, 32);
    {
        float nM = fmaxf(Mx, oM);
        float sa = __expf(Mx - nM);
        float sb = __expf(oM - nM);
        S  = S * sa + oS * sb;
        V0 = V0 * sa + o0 * sb; V1 = V1 * sa + o1 * sb;
        V2 = V2 * sa + o2 * sb; V3 = V3 * sa + o3 * sb;
    }
    if (lane < 16) {
        float inv = 1.f / S;
        float4 xi = *(const float4*)(xb + il * 4);
        float4 ob = *(const float4*)(obs + b * 4);
        float4 gv;
        gv.x = -(xi.x - A_C * (V0 * inv)) - (xi.x - ob.x);
        gv.y = -(xi.y - A_C * (V1 * inv)) - (xi.y - ob.y);
        gv.z = -(xi.z - A_C * (V2 * inv)) - (xi.z - ob.z);
        gv.w = -(xi.w - A_C * (V3 * inv)) - (xi.w - ob.w);
        *(float4*)(g + (size_t)b * Nn * 4 + il * 4) = gv;
    }
}

// ---------------- Kstein: RBF accumulate via WMMA + RMSprop update ---------
__global__ void k_stein(const float* __restrict__ x, const float* __restrict__ x2,
                        const float* __restrict__ g, const float* __restrict__ hval,
                        float* __restrict__ stepb, float* __restrict__ xn,
                        float* __restrict__ x2n) {
    int b = blockIdx.x >> 5, grp = blockIdx.x & 31;
    int lane = threadIdx.x & 31, wv = threadIdx.x >> 5;
    int itile = grp * 4 + wv;
    const float* xb = x + (size_t)b * Nn * 4;
    const float* x2b = x2 + (size_t)b * Nn;
    const float* gb = g + (size_t)b * Nn * 4;
    float invh = hval[b * 4 + 1], toh = hval[b * 4 + 2];
    int il = itile * 16 + (lane & 15);
    int ko = (lane >> 4) * 2;
    v2f bm = ld2(xb + il * 4 + ko);
    float x2i = x2b[il];
    float rsum = 0.f;
    float k0 = 0.f, k1 = 0.f, k2 = 0.f, k3 = 0.f;   // sum rbf * g_j
    float c0 = 0.f, c1 = 0.f, c2 = 0.f, c3 = 0.f;   // sum rbf * x_j
    for (int J = 0; J < NTILE; ++J) {
        int jr = J * 16 + (lane & 15);
        __builtin_prefetch(xb + (J + 1) * 64, 0, 3);
        __builtin_prefetch(gb + (J + 1) * 64, 0, 3);
        v2f am = ld2(xb + jr * 4 + ko);
        v8f d = wmma4(am, bm);                     // d[u] = x_j . x_i
        int jb = J * 16 + ((lane >> 4) << 3);
        float4 q0 = *(const float4*)(x2b + jb);
        float4 q1 = *(const float4*)(x2b + jb + 4);
        float x2j[8] = {q0.x, q0.y, q0.z, q0.w, q1.x, q1.y, q1.z, q1.w};
#pragma unroll
        for (int u = 0; u < 8; ++u) {
            int j = jb + u;
            float sq = fmaxf(x2j[u] + x2i - 2.f * d[u], 0.f);
            float r = __expf(-sq * invh);
            rsum += r;
            float4 gj = *(const float4*)(gb + j * 4);
            float4 xj = *(const float4*)(xb + j * 4);
            k0 += r * gj.x; k1 += r * gj.y; k2 += r * gj.z; k3 += r * gj.w;
            c0 += r * xj.x; c1 += r * xj.y; c2 += r * xj.z; c3 += r * xj.w;
        }
    }
    rsum += __shfl_xor(rsum, 16, 32);
    k0 += __shfl_xor(k0, 16, 32); k1 += __shfl_xor(k1, 16, 32);
    k2 += __shfl_xor(k2, 16, 32); k3 += __shfl_xor(k3, 16, 32);
    c0 += __shfl_xor(c0, 16, 32); c1 += __shfl_xor(c1, 16, 32);
    c2 += __shfl_xor(c2, 16, 32); c3 += __shfl_xor(c3, 16, 32);
    if (lane < 16) {
        float4 xi = *(const float4*)(xb + il * 4);
        const float invN = 1.f / (float)Nn;
        float s0 = (k0 + toh * (xi.x * rsum - c0)) * invN;
        float s1 = (k1 + toh * (xi.y * rsum - c1)) * invN;
        float s2 = (k2 + toh * (xi.z * rsum - c2)) * invN;
        float s3 = (k3 + toh * (xi.w * rsum - c3)) * invN;
        size_t si = (size_t)b * Nn * 4 + il * 4;
        float4 st = *(const float4*)(stepb + si);
        st.x = ALPHA_C * st.x + (1.f - ALPHA_C) * s0 * s0;
        st.y = ALPHA_C * st.y + (1.f - ALPHA_C) * s1 * s1;
        st.z = ALPHA_C * st.z + (1.f - ALPHA_C) * s2 * s2;
        st.w = ALPHA_C * st.w + (1.f - ALPHA_C) * s3 * s3;
        *(float4*)(stepb + si) = st;
        float4 xo;
        xo.x = xi.x + LR_C * s0 / (sqrtf(st.x) + EPS_C);
        xo.y = xi.y + LR_C * s1 / (sqrtf(st.y) + EPS_C);
        xo.z = xi.z + LR_C * s2 / (sqrtf(st.z) + EPS_C);
        xo.w = xi.w + LR_C * s3 / (sqrtf(st.w) + EPS_C);
        *(float4*)(xn + si) = xo;
        x2n[(size_t)b * Nn + il] = xo.x * xo.x + xo.y * xo.y + xo.z * xo.z + xo.w * xo.w;
    }
}

// ---------------- Kout: concat(new_state, -logN) ---------------------------
__global__ void k_out(const float* __restrict__ x, float* __restrict__ out) {
    int t = blockIdx.x * blockDim.x + threadIdx.x;
    if (t >= Bn * Nn) return;
    float4 xv = ((const float4*)x)[t];
    out[(size_t)t * 5 + 0] = xv.x;
    out[(size_t)t * 5 + 1] = xv.y;
    out[(size_t)t * 5 + 2] = xv.z;
    out[(size_t)t * 5 + 3] = xv.w;
    out[(size_t)t * 5 + 4] = -LOGN;
}

// ---------------- host orchestration ---------------------------------------
extern "C" void kernel_launch(void* const* d_in, const int* in_sizes, int n_in,
                              void* d_out, int out_size, void* d_ws, size_t ws_size,
                              hipStream_t stream) {
    const float* prev = (const float*)d_in[0];   // (8,2048,4)
    const float* obs  = (const float*)d_in[1];   // (8,4)
    const float* noise = (const float*)d_in[2];  // (8,2048,4)
    float* out = (float*)d_out;                  // (8,2048,5)

    float* W = (float*)d_ws;                     // ~1.28 MB used
    const size_t NBD = (size_t)Bn * Nn * 4;      // 65536
    const size_t NBn = (size_t)Bn * Nn;          // 16384
    float* x0   = W;                   // xbuf ping
    float* x1   = x0 + NBD;            // xbuf pong
    float* x2b0 = x1 + NBD;            // |x|^2 ping
    float* x2b1 = x2b0 + NBn;          // |x|^2 pong
    float* p2   = x2b1 + NBn;          // |prev|^2
    float* g    = p2 + NBn;            // posterior score
    float* stp  = g + NBD;             // RMSprop accumulator
    int*   hist = (int*)(stp + NBD);   // B x 1024 bins
    float* medinfo = (float*)(hist + Bn * NB);   // {lo, hi, rankRem, pad} x B
    float* hval = medinfo + Bn * 4;              // {h, 1/h, 2/h, pad} x B

    dim3 bN(256), gN((Bn * Nn) / 256);           // 64 blocks
    dim3 bT(128), gT(Bn * 32);                   // 256 blocks, 4 waves each
    dim3 bS(NB), gS(Bn);                         // per-batch scans

    k_init<<<gN, bN, 0, stream>>>(prev, noise, x0, x2b0, p2, stp);

    for (int it = 0; it < 3; ++it) {
        float* xc  = (it & 1) ? x1 : x0;
        float* xnx = (it & 1) ? x0 : x1;
        float* x2c = (it & 1) ? x2b1 : x2b0;
        float* x2n = (it & 1) ? x2b0 : x2b1;
        // bandwidth: two-level histogram median selection in squared-distance space
        k_prep<<<gS, bS, 0, stream>>>(x2c, medinfo, hist);
        k_hist<<<gT, bT, 0, stream>>>(xc, x2c, medinfo, hist, 0);
        k_scan<<<gS, bS, 0, stream>>>(hist, medinfo, hval, 1);
        k_hist<<<gT, bT, 0, stream>>>(xc, x2c, medinfo, hist, 1);
        k_scan<<<gS, bS, 0, stream>>>(hist, medinfo, hval, 0);
        // posterior score + Stein update
        k_grad<<<gT, bT, 0, stream>>>(xc, prev, p2, obs, g);
        k_stein<<<gT, bT, 0, stream>>>(xc, x2c, g, hval, stp, xnx, x2n);
    }
    // 3 iterations -> final state in x1 (it=2 writes pong of x0)
    k_out<<<gN, bN, 0, stream>>>(x1, out);
}